// ValueNormVec_90340342104516
// MI455X (gfx1250) — compile-verified
//
#include <hip/hip_runtime.h>
#include <hip/hip_bf16.h>

typedef float v2f __attribute__((ext_vector_type(2)));
typedef float v8f __attribute__((ext_vector_type(8)));

#define D_FEAT 128
#define EPSV 1e-5f

// ---------------------------------------------------------------------------
// Kernel 1: per-block partial column sums of x and x^2 using fp32 WMMA.
// Block = 256 threads = 8 waves; wave w owns columns [16w, 16w+16).
// Main loop: guard-free 16-row tiles, 8 unconditional b32 loads per lane and
// 8 chained V_WMMA_F32_16X16X4_F32 (A = all-ones => D[m,n] += sum_k B[k,n]).
// Tail rows (<16) folded in scalar form by block 0 after the WMMA loop.
// Partials: ws[b*256 + d] = sum_x, ws[b*256 + 128 + d] = sum_x2.
// ---------------------------------------------------------------------------
__global__ __launch_bounds__(256) void vn_stats_kernel(
    const float* __restrict__ x, float* __restrict__ partial, int N) {
  const int lane    = threadIdx.x & 31;
  const int wave    = threadIdx.x >> 5;        // 0..7
  const int halfsel = lane >> 4;               // 0: rows +0/+1, 1: rows +2/+3
  const int col     = (wave << 4) + (lane & 15);

  v2f a; a.x = 1.0f; a.y = 1.0f;               // A = ones (K-layout agnostic)
  v8f acc  = {};                               // column sums
  v8f acc2 = {};                               // column sums of squares

  const int Nt = N & ~15;                      // rows covered by 16-row tiles
  const long long step = (long long)gridDim.x * 16;

  // Lane's pointer inside the tile: +2*halfsel rows, its own column.
  const float* p = x + ((long long)blockIdx.x * 16 + 2 * halfsel) * D_FEAT + col;

  for (long long r = (long long)blockIdx.x * 16; r < (long long)Nt; r += step) {
    __builtin_prefetch(p + step * D_FEAT, 0, 1);   // global_prefetch_b8
#pragma unroll
    for (int u = 0; u < 4; ++u) {
      const float b0 = p[(4 * u + 0) * D_FEAT];    // rows r+4u+2h, r+4u+2h+1
      const float b1 = p[(4 * u + 1) * D_FEAT];
      v2f b;  b.x  = b0;      b.y  = b1;
      v2f b2; b2.x = b0 * b0; b2.y = b1 * b1;
      acc  = __builtin_amdgcn_wmma_f32_16x16x4_f32(false, a, false, b,
                                                   (short)0, acc,  false, false);
      acc2 = __builtin_amdgcn_wmma_f32_16x16x4_f32(false, a, false, b2,
                                                   (short)0, acc2, false, false);
    }
    p += step * D_FEAT;
  }

  // C/D layout: VGPR0, lanes 0-15 hold M=0, N=lane -> column sum for `col`.
  float S = acc[0];
  float Q = acc2[0];

  // Tail rows [Nt, N): accumulated once, by block 0 (lanes 0-15 only).
  if (blockIdx.x == 0 && lane < 16) {
    for (int r = Nt; r < N; ++r) {
      const float v = x[(long long)r * D_FEAT + col];
      S += v;
      Q += v * v;
    }
  }

  if (lane < 16) {
    const size_t base = (size_t)blockIdx.x * 256;
    partial[base + col]          = S;
    partial[base + D_FEAT + col] = Q;
  }
}

// ---------------------------------------------------------------------------
// Kernel 2: deterministic cross-block reduction (fixed order, no atomics) +
// Chan/Welford combine with running stats. 1 block, 128 threads (thread = d).
// stats[d] = new_mean, stats[128+d] = 1/std.
// ---------------------------------------------------------------------------
__global__ __launch_bounds__(128) void vn_finalize_kernel(
    const float* __restrict__ partial, int nblocks,
    const float* __restrict__ count, const float* __restrict__ mean,
    const float* __restrict__ m2, float* __restrict__ stats, float fN) {
  const int d = threadIdx.x;
  float S = 0.0f, Q = 0.0f;
  for (int b = 0; b < nblocks; ++b) {
    const size_t base = (size_t)b * 256;
    S += partial[base + d];
    Q += partial[base + D_FEAT + d];
  }
  const float mean_b = S / fN;
  const float m2_b   = Q - fN * mean_b * mean_b;   // sum (x - mean_b)^2

  const float cnt       = count[d];
  const float new_count = cnt + fN;
  const float delta     = mean_b - mean[d];
  const float new_mean  = mean[d] + delta * (fN / new_count);
  const float new_m2    = m2[d] + m2_b + delta * delta * (cnt * fN / new_count);
  const float denom     = fmaxf(new_count - 1.0f, 1.0f);
  const float inv_std   = 1.0f / sqrtf(new_m2 / denom + EPSV);

  stats[d]          = new_mean;
  stats[D_FEAT + d] = inv_std;
}

// ---------------------------------------------------------------------------
// Kernel 3: normalize, float4-vectorized grid-stride. Stride is a multiple of
// 32 vec4 (= one 128-col row), so each thread's column group is invariant and
// mean/inv_std are hoisted out of the loop.
// ---------------------------------------------------------------------------
__global__ __launch_bounds__(256) void vn_normalize_kernel(
    const float* __restrict__ x, float* __restrict__ out,
    const float* __restrict__ stats, long long nvec4) {
  long long idx = (long long)blockIdx.x * 256 + threadIdx.x;
  const long long stride = (long long)gridDim.x * 256;

  const int d4 = (int)(idx & 31);               // float4 slot within row
  const float4 mu = ((const float4*)stats)[d4];
  const float4 is = ((const float4*)(stats + D_FEAT))[d4];

  const float4* __restrict__ x4 = (const float4*)x;
  float4* __restrict__ o4       = (float4*)out;

  for (; idx < nvec4; idx += stride) {
    __builtin_prefetch(&x4[idx + stride], 0, 1);   // global_prefetch_b8
    float4 v = x4[idx];
    float4 o;
    o.x = (v.x - mu.x) * is.x;
    o.y = (v.y - mu.y) * is.y;
    o.z = (v.z - mu.z) * is.z;
    o.w = (v.w - mu.w) * is.w;
    o4[idx] = o;
  }
}

extern "C" void kernel_launch(void* const* d_in, const int* in_sizes, int n_in,
                              void* d_out, int out_size, void* d_ws, size_t ws_size,
                              hipStream_t stream) {
  const float* x     = (const float*)d_in[0];
  const float* count = (const float*)d_in[1];
  const float* mean  = (const float*)d_in[2];
  const float* m2    = (const float*)d_in[3];
  float* out = (float*)d_out;
  float* ws  = (float*)d_ws;

  const long long total = (long long)in_sizes[0];
  const int N = (int)(total / D_FEAT);

  // Size the partial-sum grid from the available workspace (deterministic).
  long long ws_floats = (long long)(ws_size / sizeof(float));
  long long cap = (ws_floats - 2 * D_FEAT) / 256;   // slots of 256 floats each
  if (cap < 1) cap = 1;
  long long nb = cap < 2048 ? cap : 2048;
  long long tiles = ((long long)N + 15) / 16;       // no more blocks than tiles
  if (tiles < 1) tiles = 1;
  if (nb > tiles) nb = tiles;
  const int nblocks = (int)nb;

  float* partial = ws;                              // nblocks * 256 floats
  float* stats   = ws + (size_t)nblocks * 256;      // 256 floats: mean | inv_std

  vn_stats_kernel<<<nblocks, 256, 0, stream>>>(x, partial, N);
  vn_finalize_kernel<<<1, D_FEAT, 0, stream>>>(partial, nblocks, count, mean, m2,
                                               stats, (float)N);

  const long long nvec4 = total / 4;                // N*32 float4's
  long long g3 = (nvec4 + 255) / 256;
  if (g3 > 32768) g3 = 32768;
  if (g3 < 1) g3 = 1;
  vn_normalize_kernel<<<(unsigned)g3, 256, 0, stream>>>(x, out, stats, nvec4);
}